// SmoothnessLoss_24249385353751
// MI455X (gfx1250) — compile-verified
//
#include <hip/hip_runtime.h>
#include <hip/hip_bf16.h>

typedef float v2f __attribute__((ext_vector_type(2)));
typedef float v8f __attribute__((ext_vector_type(8)));

#define BATCH 8
#define NPTS 4096
#define NSAMP 32
#define R2 0.25f
#define WAVES_PER_BLOCK 8
#define ROWTILES_PER_BATCH (NPTS / 16)                 /* 256 */
#define TOTAL_WAVES (BATCH * ROWTILES_PER_BATCH)       /* 2048 */
#define NBLOCKS (TOTAL_WAVES / WAVES_PER_BLOCK)        /* 256 */

// One wave handles 16 query rows of one batch. Pairwise distances are produced
// 16x16 at a time with V_WMMA_F32_16X16X4_F32:
//   A[M][K] = (x_i, y_i, z_i, 1)        (16x4)
//   B[K][N] = (-2x_j, -2y_j, -2z_j, sq_j)  (4x16)
//   D[M][N] = sq_j - 2*p_i.p_j   ;  hit  <=>  D < 0.25 - sq_i
__global__ void __launch_bounds__(256)
ball_query_loss_kernel(const float* __restrict__ pc1,
                       const float* __restrict__ flow,
                       float* __restrict__ partials) {
    const int lane = threadIdx.x & 31;
    const int wid  = threadIdx.x >> 5;
    const int gw   = blockIdx.x * WAVES_PER_BLOCK + wid;
    const int b    = gw / ROWTILES_PER_BATCH;
    const int rt   = gw % ROWTILES_PER_BATCH;
    const int rowbase = rt * 16;

    const float* __restrict__ px = pc1 + (size_t)(b * 3 + 0) * NPTS;
    const float* __restrict__ py = pc1 + (size_t)(b * 3 + 1) * NPTS;
    const float* __restrict__ pz = pc1 + (size_t)(b * 3 + 2) * NPTS;
    const float* __restrict__ fx = flow + (size_t)(b * 3 + 0) * NPTS;
    const float* __restrict__ fy = flow + (size_t)(b * 3 + 1) * NPTS;
    const float* __restrict__ fz = flow + (size_t)(b * 3 + 2) * NPTS;

    const int  l16 = lane & 15;
    const bool lo  = lane < 16;

    // ---- A fragment (constant across the whole j scan) ----
    // ISA layout (32-bit A 16x4): VGPR0 = K0 (lanes 0-15) / K2 (lanes 16-31),
    //                             VGPR1 = K1 / K3.   M = lane & 15.
    const int i = rowbase + l16;
    const float xi = px[i], yi = py[i], zi = pz[i];
    v2f afrag;
    afrag[0] = lo ? xi : zi;
    afrag[1] = lo ? yi : 1.0f;

    // per-row threshold t[M] = r^2 - |p_i|^2 ; row M's value lives in lane M
    const float tl = R2 - (xi * xi + yi * yi + zi * zi);
    float tv[8];
#pragma unroll
    for (int v = 0; v < 8; ++v)
        tv[v] = __shfl(tl, lo ? v : (v + 8));   // lane needs t[v + (hi?8:0)]

    // query flow (owner lane m<16 owns row rowbase+m)
    const float fxi = fx[i], fyi = fy[i], fzi = fz[i];

    int   cnt    = 0;
    int   firstj = i;       // fallback; j==i always passes so this gets set
    float acc    = 0.0f;
    const v8f czero = {};

    for (int tb = 0; tb < NPTS; tb += 16) {
        // ---- B fragment for this j tile (both half-waves load same 16 j) ----
        const int j = tb + l16;
        const float xj = px[j], yj = py[j], zj = pz[j];
        v2f bfrag;
        bfrag[0] = lo ? (-2.0f * xj) : (-2.0f * zj);                   // K0 / K2
        bfrag[1] = lo ? (-2.0f * yj) : (xj * xj + yj * yj + zj * zj);  // K1 / K3

        const v8f d = __builtin_amdgcn_wmma_f32_16x16x4_f32(
            /*neg_a=*/false, afrag, /*neg_b=*/false, bfrag,
            /*c_mod=*/(short)0, czero, /*reuse_a=*/false, /*reuse_b=*/false);

        // D layout: VGPR v -> row v (lanes 0-15, N=lane) and row v+8
        // (lanes 16-31, N=lane-16). One ballot per VGPR packs two rows'
        // 16 hit bits, already ordered by j within the tile.
        unsigned mask[8];
#pragma unroll
        for (int v = 0; v < 8; ++v)
            mask[v] = (unsigned)__ballot(d[v] < tv[v]);

        if (lo) {
            unsigned mm = 0u;
#pragma unroll
            for (int v = 0; v < 8; ++v) {
                if (lane == v)     mm = mask[v] & 0xFFFFu;
                if (lane == v + 8) mm = mask[v] >> 16;
            }
            // CUDA ball-query semantics: first NSAMP hits in index order
            while (mm && cnt < NSAMP) {
                const int n = __ffs(mm) - 1;
                mm &= mm - 1u;
                const int jj = tb + n;
                if (cnt == 0) firstj = jj;
                const float dx = fx[jj] - fxi;
                const float dy = fy[jj] - fyi;
                const float dz = fz[jj] - fzi;
                const float s  = dx * dx + dy * dy + dz * dz;
                acc += (s > 0.0f) ? __builtin_sqrtf(s) : 0.0f;
                ++cnt;
            }
        }

        // whole-wave early exit once all 16 rows are full
        const bool full = (!lo) || (cnt >= NSAMP);
        if (__all(full)) break;
    }

    // unfilled slots repeat the first hit
    if (lo) {
        const float dx = fx[firstj] - fxi;
        const float dy = fy[firstj] - fyi;
        const float dz = fz[firstj] - fzi;
        const float s  = dx * dx + dy * dy + dz * dz;
        const float nr = (s > 0.0f) ? __builtin_sqrtf(s) : 0.0f;
        acc += (float)(NSAMP - cnt) * nr;
    } else {
        acc = 0.0f;
    }

    // wave32 reduction, lane 0 writes the wave partial
#pragma unroll
    for (int off = 16; off >= 1; off >>= 1)
        acc += __shfl_down(acc, off);
    if (lane == 0) partials[gw] = acc;
}

__global__ void __launch_bounds__(256)
reduce_partials_kernel(const float* __restrict__ partials, float* __restrict__ out) {
    __shared__ float sm[256];
    float s = 0.0f;
    for (int idx = threadIdx.x; idx < TOTAL_WAVES; idx += 256)
        s += partials[idx];
    sm[threadIdx.x] = s;
    __syncthreads();
    for (int off = 128; off >= 1; off >>= 1) {
        if ((int)threadIdx.x < off) sm[threadIdx.x] += sm[threadIdx.x + off];
        __syncthreads();
    }
    if (threadIdx.x == 0)
        out[0] = sm[0] * (1.0f / (float)(BATCH * NSAMP));
}

extern "C" void kernel_launch(void* const* d_in, const int* in_sizes, int n_in,
                              void* d_out, int out_size, void* d_ws, size_t ws_size,
                              hipStream_t stream) {
    const float* pc1  = (const float*)d_in[0];
    const float* flow = (const float*)d_in[1];
    float* partials   = (float*)d_ws;            // 2048 floats = 8 KB
    float* out        = (float*)d_out;

    ball_query_loss_kernel<<<NBLOCKS, 256, 0, stream>>>(pc1, flow, partials);
    reduce_partials_kernel<<<1, 256, 0, stream>>>(partials, out);
}